// Gated_24592982736976
// MI455X (gfx1250) — compile-verified
//
#include <hip/hip_runtime.h>

typedef float v2f __attribute__((ext_vector_type(2)));
typedef float v4f __attribute__((ext_vector_type(4)));
typedef float v8f __attribute__((ext_vector_type(8)));

#define D_DIM 256

// ---- monotone float <-> uint mapping for atomic max on f32 ----
__device__ __forceinline__ unsigned fmap(float f) {
  unsigned b = __float_as_uint(f);
  return (b & 0x80000000u) ? ~b : (b | 0x80000000u);
}
__device__ __forceinline__ float funmap(unsigned m) {
  return __uint_as_float((m & 0x80000000u) ? (m ^ 0x80000000u) : ~m);
}

// K0: zero H, smax bits, denom
__global__ void gp_init(float* __restrict__ H, int outN,
                        unsigned* __restrict__ smax, float* __restrict__ denom, int nseg) {
  int i = blockIdx.x * blockDim.x + threadIdx.x;
  int stride = gridDim.x * blockDim.x;
  for (int t = i; t < outN; t += stride) H[t] = 0.0f;
  for (int t = i; t < nseg; t += stride) { smax[t] = 0u; denom[t] = 0.0f; }
}

// K1: per-node score = dot(x[i], W) + b ; atomic segment max. One wave per node.
__global__ void gp_scores(const float* __restrict__ x, const float* __restrict__ W,
                          const float* __restrict__ bias, const long long* __restrict__ batch,
                          float* __restrict__ scores, unsigned* __restrict__ smax, int N) {
  const int lane = threadIdx.x & 31;
  const int wid  = (blockIdx.x * blockDim.x + threadIdx.x) >> 5;
  const int nw   = (gridDim.x * blockDim.x) >> 5;
  const float b0 = bias[0];
  const v4f* w4 = (const v4f*)W;
  const v4f wa = w4[lane * 2 + 0];
  const v4f wb = w4[lane * 2 + 1];
  for (int node = wid; node < N; node += nw) {
    const v4f* r = (const v4f*)(x + (size_t)node * D_DIM);
    v4f xa = __builtin_nontemporal_load(&r[lane * 2 + 0]);   // x is single-use stream
    v4f xb = __builtin_nontemporal_load(&r[lane * 2 + 1]);
    float p = xa.x * wa.x + xa.y * wa.y + xa.z * wa.z + xa.w * wa.w +
              xb.x * wb.x + xb.y * wb.y + xb.z * wb.z + xb.w * wb.w;
#pragma unroll
    for (int off = 16; off > 0; off >>= 1) p += __shfl_xor(p, off, 32);
    if (lane == 0) {
      float s = p + b0;
      scores[node] = s;
      atomicMax(&smax[(int)batch[node]], fmap(s));
    }
  }
}

// K2: ex = exp(score - smax[seg]) in place; denom[seg] += ex (native f32 atomic add)
__global__ void gp_exp(const long long* __restrict__ batch, const unsigned* __restrict__ smax,
                       float* __restrict__ scores_ex, float* __restrict__ denom, int N) {
  int i = blockIdx.x * blockDim.x + threadIdx.x;
  int stride = gridDim.x * blockDim.x;
  for (; i < N; i += stride) {
    int seg = (int)batch[i];
    float e = __expf(scores_ex[i] - funmap(smax[seg]));
    scores_ex[i] = e;
    atomicAdd(&denom[seg], e);
  }
}

// K3: segment start offsets via binary search on the sorted index (start[nseg] = N)
__global__ void gp_bounds(const long long* __restrict__ batch, int* __restrict__ start,
                          int N, int nseg) {
  int s = blockIdx.x * blockDim.x + threadIdx.x;
  if (s > nseg) return;
  int lo = 0, hi = N;
  while (lo < hi) { int mid = (lo + hi) >> 1; if ((int)batch[mid] < (long long)s) lo = mid + 1; else hi = mid; }
  start[s] = lo;
}

// K4: H[s,:] = sum_i alpha_i * x[i,:] as block-sparse WMMA f32 16x16x4.
// Block = 16 consecutive segments x (grid.y split of the node range).
// Wave w owns dim-chunks 2w and 2w+1 (16 dims each); 8 waves cover D=256.
// A-mask built from per-lane segment bounds (sorted index) -> branch-free inner loop.
__global__ void gp_pool(const float* __restrict__ x, const float* __restrict__ ex,
                        const float* __restrict__ denom, const int* __restrict__ start,
                        float* __restrict__ H, int nseg) {
  const int lane = threadIdx.x & 31;
  const int wave = threadIdx.x >> 5;
  const int m    = lane & 15;              // fragment row (A) / col (B,D)
  const int kOff = (lane >> 4) << 1;       // lanes 0-15 -> K{0,1}, lanes 16-31 -> K{2,3}

  const int segLo = (int)blockIdx.x << 4;
  const int sBeg  = start[segLo];
  const int sEnd  = start[min(segLo + 16, nseg)];
  const int total = sEnd - sBeg;
  if (total <= 0) return;                      // uniform across block
  const int per = (total + (int)gridDim.y - 1) / (int)gridDim.y;
  const int lo  = sBeg + per * (int)blockIdx.y;
  const int hi  = min(sEnd, lo + per);
  if (lo >= hi) return;                        // uniform across block

  // per-lane segment: softmax denominator + node range [bm, em)
  const int myseg = min(segLo + m, nseg - 1);
  const float dn  = denom[myseg];
  const float inv = (dn > 0.0f) ? (1.0f / dn) : 0.0f;
  const int bm = start[myseg];
  const int em = min(start[myseg + 1], hi);
  const unsigned len = (unsigned)max(0, em - bm);   // pred: (unsigned)(n-bm) < len

  const int c0 = wave * 2;                     // two 16-dim chunks per wave
  v8f acc0 = {};
  v8f acc1 = {};

  const int cnt   = hi - lo;
  const int nmain = cnt & ~3;

  int n = lo + kOff;                           // this lane-half's node cursor
  const float* xp = x + (size_t)n * D_DIM + c0 * 16 + m;
  const float* ep = ex + n;

#pragma unroll 2
  for (int it = 0; it < nmain; it += 4) {
    float e0  = ep[0];
    float e1  = ep[1];
    float b0x = __builtin_nontemporal_load(xp);
    float b1x = __builtin_nontemporal_load(xp + 16);
    float b0y = __builtin_nontemporal_load(xp + D_DIM);
    float b1y = __builtin_nontemporal_load(xp + D_DIM + 16);
    const unsigned u0 = (unsigned)(n - bm);
    const unsigned u1 = (unsigned)(n + 1 - bm);
    v2f A;  A.x = (u0 < len) ? e0 * inv : 0.0f;   // A[m][kOff], A[m][kOff+1]
            A.y = (u1 < len) ? e1 * inv : 0.0f;
    v2f B0; B0.x = b0x; B0.y = b0y;               // B[kOff][n], B[kOff+1][n] chunk c0
    v2f B1; B1.x = b1x; B1.y = b1y;               // chunk c0+1
    acc0 = __builtin_amdgcn_wmma_f32_16x16x4_f32(false, A, false, B0, (short)0, acc0, false, false);
    acc1 = __builtin_amdgcn_wmma_f32_16x16x4_f32(false, A, false, B1, (short)0, acc1, false, false);
    n += 4; xp += 4 * D_DIM; ep += 4;
  }

  if (nmain < cnt) {                           // uniform; EXEC stays full
    // tail: clamp addresses to last valid node, mask invalid k-slots via predicate
    const int n0 = min(n, hi - 1);
    const int n1 = min(n + 1, hi - 1);
    const float* r0 = x + (size_t)n0 * D_DIM + c0 * 16 + m;
    const float* r1 = x + (size_t)n1 * D_DIM + c0 * 16 + m;
    const float e0 = ex[n0];
    const float e1 = ex[n1];
    const bool p0 = ((unsigned)(n - bm) < len) && (n < hi);
    const bool p1 = ((unsigned)(n + 1 - bm) < len) && (n + 1 < hi);
    v2f A;  A.x = p0 ? e0 * inv : 0.0f;
            A.y = p1 ? e1 * inv : 0.0f;
    v2f B0; B0.x = r0[0];  B0.y = r1[0];
    v2f B1; B1.x = r0[16]; B1.y = r1[16];
    acc0 = __builtin_amdgcn_wmma_f32_16x16x4_f32(false, A, false, B0, (short)0, acc0, false, false);
    acc1 = __builtin_amdgcn_wmma_f32_16x16x4_f32(false, A, false, B1, (short)0, acc1, false, false);
  }

  // D layout: VGPR v at lane l = D[v + 8*(l>=16)][l%16]
  const int rbase = segLo + ((lane >> 4) << 3);
#pragma unroll
  for (int v = 0; v < 8; ++v) {
    const int row = rbase + v;
    if (row < nseg) {
      atomicAdd(&H[(size_t)row * D_DIM + c0 * 16 + m], acc0[v]);
      atomicAdd(&H[(size_t)row * D_DIM + (c0 + 1) * 16 + m], acc1[v]);
    }
  }
}

extern "C" void kernel_launch(void* const* d_in, const int* in_sizes, int n_in,
                              void* d_out, int out_size, void* d_ws, size_t ws_size,
                              hipStream_t stream) {
  const float*     x     = (const float*)d_in[0];
  const float*     W     = (const float*)d_in[1];
  const float*     bias  = (const float*)d_in[2];
  const long long* batch = (const long long*)d_in[3];
  float* H = (float*)d_out;

  const int N    = in_sizes[0] / D_DIM;   // 1,000,000
  const int NSEG = out_size / D_DIM;      // 4096

  // workspace layout
  float*    scores = (float*)d_ws;            // N floats (becomes ex in place)
  unsigned* smax   = (unsigned*)(scores + N); // NSEG
  float*    denom  = (float*)(smax + NSEG);   // NSEG
  int*      startv = (int*)(denom + NSEG);    // NSEG + 1

  // K0: init output + segment accumulators
  gp_init<<<1024, 256, 0, stream>>>(H, out_size, smax, denom, NSEG);

  // K1: scores + segment max (wave per node)
  gp_scores<<<1024, 256, 0, stream>>>(x, W, bias, batch, scores, smax, N);

  // K2: exp + denominator
  gp_exp<<<1024, 256, 0, stream>>>(batch, smax, scores, denom, N);

  // K3: segment boundaries
  gp_bounds<<<(NSEG + 1 + 255) / 256, 256, 0, stream>>>(batch, startv, N, NSEG);

  // K4: WMMA block-sparse weighted pooling
  const int SPLIT = 4;
  dim3 grid((NSEG + 15) / 16, SPLIT);
  gp_pool<<<grid, 256, 0, stream>>>(x, scores, denom, startv, H, NSEG);
}